// AMASEQC_multirun_36112085025352
// MI455X (gfx1250) — compile-verified
//
#include <hip/hip_runtime.h>

typedef __attribute__((ext_vector_type(2))) float v2f;
typedef __attribute__((ext_vector_type(8))) float v8f;

#define P_DIM 32
#define E_DIM 4
#define R_DIM 8192
#define C_DIM 8192
#define NREF  (E_DIM * R_DIM)   // 32768
#define NATM  (E_DIM * C_DIM)   // 32768
#define KSTEPS (R_DIM / 4)      // 2048
#define PF_DIST 32              // prefetch distance in K-steps (~720ns ahead)

// ---------------------------------------------------------------------------
// Kernel 1: gather Alpha through ref_idx and pack it into per-lane WMMA
// A-fragment order:  ws[((e*KSTEPS + k)*32 + lane)] = float4{
//    Ag[4k+t][p], Ag[4k+t+1][p], Ag[4k+t][p+16], Ag[4k+t+1][p+16] }
// with t = 2*(lane>=16), p = lane&15, Ag[r][p] = Alpha[p][ref_idx[e][r]].
// Matches the ISA 32-bit A 16x4 layout (lanes 0-15: K=0/1, lanes 16-31:
// K=2/3), so the GEMM loads A with one b128 per K-step.
// ---------------------------------------------------------------------------
__global__ __launch_bounds__(256) void gather_pack_kernel(
    const float* __restrict__ Alpha,
    const long long* __restrict__ ref_idx,
    float* __restrict__ wsA) {
  int tid  = blockIdx.x * blockDim.x + threadIdx.x;  // 0 .. E*KSTEPS*32-1
  int lane = tid & 31;
  int k    = (tid >> 5) & (KSTEPS - 1);
  int e    = tid >> 16;                              // KSTEPS*32 == 65536
  int t    = (lane >= 16) ? 2 : 0;
  int p    = lane & 15;
  int r    = 4 * k + t;
  long long c0 = ref_idx[(size_t)e * R_DIM + r];
  long long c1 = ref_idx[(size_t)e * R_DIM + r + 1];
  float4 v;
  v.x = Alpha[(size_t)p * NREF + c0];
  v.y = Alpha[(size_t)p * NREF + c1];
  v.z = Alpha[(size_t)(p + 16) * NREF + c0];
  v.w = Alpha[(size_t)(p + 16) * NREF + c1];
  ((float4*)wsA)[tid] = v;
}

// ---------------------------------------------------------------------------
// Kernel 2: per wave: 32 rows x 32 cols output tile of one expert,
// full-R accumulation with V_WMMA_F32_16X16X4_F32, scatter epilogue with p0.
// 256 blocks x 128 threads (4 waves) = 1024 waves; 64 blocks per expert.
// ---------------------------------------------------------------------------
__global__ __launch_bounds__(128) void gemm_scatter_kernel(
    const float* __restrict__ wsA,
    const float* __restrict__ Kmat,
    const float* __restrict__ p0,
    const long long* __restrict__ elm_idx,
    float* __restrict__ out) {
  const int lane = threadIdx.x & 31;
  const int wave = threadIdx.x >> 5;
  const int bx   = blockIdx.x;
  const int e    = bx >> 6;                           // 64 blocks / expert
  const int colbase = ((bx & 63) << 7) + (wave << 5); // 128 cols/block, 32/wave

  const int t   = (lane >= 16) ? 2 : 0;               // B K-row pair select
  const int nlo = lane & 15;                          // column within group

  const float* __restrict__ Ke = Kmat + (size_t)e * R_DIM * C_DIM;
  const size_t boff0 = (size_t)t * C_DIM + (size_t)colbase + nlo; // col group 0
  const size_t boff1 = boff0 + 16;                                // col group 1
  const float4* __restrict__ Ap =
      (const float4*)wsA + (size_t)e * KSTEPS * 32 + lane;

  v8f acc00 = {}; // M-tile 0 (rows 0-15),  cols colbase+0..15
  v8f acc01 = {}; // M-tile 0,              cols colbase+16..31
  v8f acc10 = {}; // M-tile 1 (rows 16-31), cols colbase+0..15
  v8f acc11 = {}; // M-tile 1,              cols colbase+16..31

#pragma unroll 8
  for (int k = 0; k < KSTEPS; ++k) {
    float4 a4 = Ap[(size_t)k * 32];                   // one b128: both A frags
    const float* Kr = Ke + (size_t)(4 * k) * C_DIM;
    float b00 = Kr[boff0];
    float b01 = Kr[boff0 + C_DIM];
    float b10 = Kr[boff1];
    float b11 = Kr[boff1 + C_DIM];

    // Stream-ahead prefetch, PF_DIST K-steps (~720ns) ahead -- enough for the
    // prefetch to complete from HBM before the demand load arrives. Lanes
    // 0-15 cover row 4kp (+1), lanes 16-31 row 4kp+2 (+3); both 16-col
    // groups share one 128B line, so two prefetches cover the whole upcoming
    // 4x32 B tile (4 cachelines -> 4 deduped requests).
    int kp = k + PF_DIST; if (kp > KSTEPS - 1) kp = KSTEPS - 1;
    const float* Kpf = Ke + (size_t)(4 * kp) * C_DIM + boff0;
    __builtin_prefetch(Kpf, 0, 3);                    // rows 4kp / 4kp+2
    __builtin_prefetch(Kpf + C_DIM, 0, 3);            // rows 4kp+1 / 4kp+3

    v2f a0  = {a4.x, a4.y};   // rows p,     K = 4k+t .. 4k+t+1
    v2f a1  = {a4.z, a4.w};   // rows p+16
    v2f bb0 = {b00, b01};     // cols +0..15
    v2f bb1 = {b10, b11};     // cols +16..31

    acc00 = __builtin_amdgcn_wmma_f32_16x16x4_f32(false, a0, false, bb0,
                                                  (short)0, acc00, false, false);
    acc10 = __builtin_amdgcn_wmma_f32_16x16x4_f32(false, a1, false, bb0,
                                                  (short)0, acc10, false, false);
    acc01 = __builtin_amdgcn_wmma_f32_16x16x4_f32(false, a0, false, bb1,
                                                  (short)0, acc01, false, false);
    acc11 = __builtin_amdgcn_wmma_f32_16x16x4_f32(false, a1, false, bb1,
                                                  (short)0, acc11, false, false);
  }

  // Scatter epilogue: C/D layout => VGPR j: rows j (lanes 0-15) / j+8
  // (lanes 16-31), N = lane&15. Each lane owns 2 scatter targets.
  const int ccol = colbase + nlo;
  const long long tg0 = elm_idx[(size_t)e * C_DIM + ccol];
  const long long tg1 = elm_idx[(size_t)e * C_DIM + ccol + 16];
  const int rb = (lane >= 16) ? 8 : 0;
#pragma unroll
  for (int j = 0; j < 8; ++j) {
    const size_t r0 = (size_t)(rb + j) * NATM;        // M-tile 0 row
    const size_t r1 = (size_t)(16 + rb + j) * NATM;   // M-tile 1 row
    out[r0 + tg0] = acc00[j] + p0[r0 + tg0];
    out[r0 + tg1] = acc01[j] + p0[r0 + tg1];
    out[r1 + tg0] = acc10[j] + p0[r1 + tg0];
    out[r1 + tg1] = acc11[j] + p0[r1 + tg1];
  }
}

extern "C" void kernel_launch(void* const* d_in, const int* in_sizes, int n_in,
                              void* d_out, int out_size, void* d_ws, size_t ws_size,
                              hipStream_t stream) {
  const float*     Alpha   = (const float*)d_in[0];      // (32, 32768) f32
  const float*     Kmat    = (const float*)d_in[1];      // (4, 8192, 8192) f32
  const float*     p0      = (const float*)d_in[2];      // (32, 32768) f32
  const long long* ref_idx = (const long long*)d_in[3];  // (4, 8192) i64
  const long long* elm_idx = (const long long*)d_in[4];  // (4, 8192) i64
  float* out = (float*)d_out;                            // (32, 32768) f32
  float* wsA = (float*)d_ws;                             // 4 MiB A-fragments

  // 1) gather + pack Alpha into WMMA A-fragment layout (E*KSTEPS*32 lanes)
  hipLaunchKernelGGL(gather_pack_kernel, dim3((E_DIM * KSTEPS * 32) / 256),
                     dim3(256), 0, stream, Alpha, ref_idx, wsA);

  // 2) HBM-rate K stream + WMMA accumulate + scatter epilogue
  hipLaunchKernelGGL(gemm_scatter_kernel, dim3(256), dim3(128), 0, stream,
                     wsA, Kmat, p0, elm_idx, out);
}